// Gemma4TextRouter_5617817223267
// MI455X (gfx1250) — compile-verified
//
#include <hip/hip_runtime.h>
#include <math.h>

#define HID 2048
#define NEXP 64
#define NTOK 16384
#define NTILE 1024          // NTOK / 16
#define WAVES 8             // waves (16-token tiles) per block
#define K4CNT (HID / 4)     // 512 k-steps of 4

typedef float v2f __attribute__((ext_vector_type(2)));
typedef float v8f __attribute__((ext_vector_type(8)));

// ---------------------------------------------------------------------------
// Kernel 1: fold scale[h] * H^-0.5 into proj_w and repack into the exact
// per-lane B-fragment layout of V_WMMA_F32_16X16X4_F32:
//   flat float2 index = ((k4*4 + ntile)*32 + lane)
//   lane<16 : { w[n0+lane][k0+0], w[n0+lane][k0+1] }
//   lane>=16: { w[n0+lane-16][k0+2], w[n0+lane-16][k0+3] }
// Total: 512 KB in d_ws — L2-resident, reused by all 1024 tiles.
// ---------------------------------------------------------------------------
__global__ __launch_bounds__(256) void fold_pack_kernel(
    const float* __restrict__ proj_w, const float* __restrict__ scale,
    float* __restrict__ bpack) {
  int i = blockIdx.x * blockDim.x + threadIdx.x;      // 65536 float2 slots
  if (i >= K4CNT * 4 * 32) return;
  int lane = i & 31;
  int nt   = (i >> 5) & 3;
  int k4   = i >> 7;
  int n = nt * 16 + (lane & 15);
  int k = k4 * 4 + ((lane >> 4) << 1);
  const float rootH = 0.02209708691207961f;           // 2048^-0.5
  v2f o;
  o.x = proj_w[n * HID + k + 0] * scale[k + 0] * rootH;
  o.y = proj_w[n * HID + k + 1] * scale[k + 1] * rootH;
  ((v2f*)bpack)[i] = o;
}

// ---------------------------------------------------------------------------
// Kernel 2: fused stream pass. One wave32 = one 16-token tile.
//   - single read of x, WMMA fp32 accumulation of 16x64 scores + sum(x^2)
//   - rsqrt scaling, softmax, top-4, renorm, per-expert scale — all fused
// ---------------------------------------------------------------------------
__global__ __launch_bounds__(32 * WAVES) void router_fused_kernel(
    const float* __restrict__ hs, const float* __restrict__ bpack,
    const float* __restrict__ per_expert_scale, float* __restrict__ out) {
  __shared__ float sc[WAVES][16][68];   // 68-stride padding: conflict-free

  const int lane = threadIdx.x & 31;
  const int w    = threadIdx.x >> 5;
  const int tile = blockIdx.x * WAVES + w;
  const int token0 = tile * 16;

  const int row   = lane & 15;
  const int khalf = (lane >> 4) << 1;                 // 0 or 2
  const float* aPtr = hs + (size_t)(token0 + row) * HID + khalf;

  v8f acc0 = {}, acc1 = {}, acc2 = {}, acc3 = {};
  float ssq = 0.0f;

  const v2f* __restrict__ bp2 = (const v2f*)bpack;

#pragma unroll 2
  for (int k4 = 0; k4 < K4CNT; ++k4) {
    v2f a = *(const v2f*)(aPtr + k4 * 4);
    ssq = fmaf(a.x, a.x, ssq);
    ssq = fmaf(a.y, a.y, ssq);
    const v2f* bp = bp2 + (size_t)k4 * 128 + lane;
    v2f b0 = bp[0];
    v2f b1 = bp[32];
    v2f b2 = bp[64];
    v2f b3 = bp[96];
    acc0 = __builtin_amdgcn_wmma_f32_16x16x4_f32(false, a, false, b0, (short)0, acc0, false, false);
    acc1 = __builtin_amdgcn_wmma_f32_16x16x4_f32(false, a, false, b1, (short)0, acc1, false, false);
    acc2 = __builtin_amdgcn_wmma_f32_16x16x4_f32(false, a, false, b2, (short)0, acc2, false, false);
    acc3 = __builtin_amdgcn_wmma_f32_16x16x4_f32(false, a, false, b3, (short)0, acc3, false, false);
  }

  // lane L (<16) holds k≡{0,1} mod 4 of row L; lane L+16 holds k≡{2,3}.
  ssq += __shfl_xor(ssq, 16);
  const float rs = rsqrtf(ssq * (1.0f / (float)HID) + 1e-6f);

  // Spill D fragments: VGPR r, lanes 0-15 -> M=r; lanes 16-31 -> M=r+8.
  const int m0  = (lane >> 4) * 8;
  const int col = lane & 15;
#pragma unroll
  for (int r = 0; r < 8; ++r) {
    sc[w][m0 + r][ 0 + col] = acc0[r];
    sc[w][m0 + r][16 + col] = acc1[r];
    sc[w][m0 + r][32 + col] = acc2[r];
    sc[w][m0 + r][48 + col] = acc3[r];
  }
  __syncthreads();

  if (lane < 16) {
    const int token = token0 + lane;
    float* srow = &sc[w][lane][0];

    float mx = -INFINITY;
#pragma unroll
    for (int e = 0; e < NEXP; ++e) {
      float v = srow[e] * rs;                         // apply RMSNorm scalar
      srow[e] = v;
      mx = fmaxf(mx, v);
    }
    float sum = 0.0f;
#pragma unroll
    for (int e = 0; e < NEXP; ++e) {
      float ex = __expf(srow[e] - mx);
      srow[e] = ex;
      sum += ex;
    }
    const float inv = 1.0f / sum;

    float* out_rp = out + (size_t)token * NEXP;
    float v0 = -1.f, v1 = -1.f, v2 = -1.f, v3 = -1.f;
    int   i0 = 0,   i1 = 0,   i2 = 0,   i3 = 0;
#pragma unroll
    for (int e = 0; e < NEXP; ++e) {
      float p = srow[e] * inv;
      out_rp[e] = p;
      // strict '>' keeps lowest index on ties (matches jax.lax.top_k)
      if (p > v0)      { v3=v2;i3=i2; v2=v1;i2=i1; v1=v0;i1=i0; v0=p;i0=e; }
      else if (p > v1) { v3=v2;i3=i2; v2=v1;i2=i1; v1=p;i1=e; }
      else if (p > v2) { v3=v2;i3=i2; v2=p;i2=e; }
      else if (p > v3) { v3=p;i3=e; }
    }
    const float invw = 1.0f / (v0 + v1 + v2 + v3);
    float* out_w = out + (size_t)NTOK * NEXP + (size_t)token * 4;
    float* out_i = out + (size_t)NTOK * NEXP + (size_t)NTOK * 4 + (size_t)token * 4;
    out_w[0] = v0 * invw * per_expert_scale[i0];
    out_w[1] = v1 * invw * per_expert_scale[i1];
    out_w[2] = v2 * invw * per_expert_scale[i2];
    out_w[3] = v3 * invw * per_expert_scale[i3];
    out_i[0] = (float)i0;
    out_i[1] = (float)i1;
    out_i[2] = (float)i2;
    out_i[3] = (float)i3;
  }
}

extern "C" void kernel_launch(void* const* d_in, const int* in_sizes, int n_in,
                              void* d_out, int out_size, void* d_ws, size_t ws_size,
                              hipStream_t stream) {
  const float* hidden_states    = (const float*)d_in[0];   // [4,4096,2048]
  const float* proj_w           = (const float*)d_in[1];   // [64,2048]
  const float* scale            = (const float*)d_in[2];   // [2048]
  const float* per_expert_scale = (const float*)d_in[3];   // [64]
  float* out   = (float*)d_out;
  float* bpack = (float*)d_ws;                             // 512 KB packed B

  (void)in_sizes; (void)n_in; (void)out_size; (void)ws_size;

  const int packSlots = K4CNT * 4 * 32;                    // 65536 float2
  fold_pack_kernel<<<(packSlots + 255) / 256, 256, 0, stream>>>(
      proj_w, scale, bpack);

  router_fused_kernel<<<NTILE / WAVES, 32 * WAVES, 0, stream>>>(
      hidden_states, bpack, per_expert_scale, out);
}